// MelGeneralizedCepstralAnalysis_79190607004296
// MI455X (gfx1250) — compile-verified
//
#include <hip/hip_runtime.h>
#include <math.h>

#define MM    24
#define LL    512
#define NH1   257
#define HPAD  272          // 17*16, padded bin count
#define NP    49           // 2M+1
#define NPAD  64
#define NR    25           // M+1
#define NRPAD 32
#define AL    0.42f
#define GAM   (-0.5f)
#define NIT   6

typedef float v2f __attribute__((ext_vector_type(2)));
typedef float v8f __attribute__((ext_vector_type(8)));

// ---------------- workspace layout (float offsets) ----------------
#define OFF_AC   0                         // 512*25
#define OFF_AP   (OFF_AC + LL*(MM+1))      // 49*512
#define OFF_AR   (OFF_AP + NP*LL)          // 25*512
#define OFF_TABC (OFF_AR + NR*LL)          // 512
#define OFF_TABS (OFF_TABC + LL)           // 512
#define OFF_MR   (OFF_TABS + LL)           // 24*272
#define OFF_MI   (OFF_MR + MM*HPAD)
#define OFF_PR   (OFF_MI + MM*HPAD)        // 272*64
#define OFF_PI   (OFF_PR + HPAD*NPAD)
#define OFF_RR   (OFF_PI + HPAD*NPAD)      // 272*32
#define OFF_RI   (OFF_RR + HPAD*NRPAD)
#define CONST_FLOATS (2*MM*HPAD + 2*HPAD*NPAD + 2*HPAD*NRPAD)  // hot matrices

// ---------------- WMMA helpers ----------------
__device__ __forceinline__ v8f wmma4(v2f a, v2f b, v8f c) {
  // D(16x16 f32) = A(16x4 f32) * B(4x16 f32) + C
  return __builtin_amdgcn_wmma_f32_16x16x4_f32(false, a, false, b, (short)0, c,
                                               false, false);
}
__device__ __forceinline__ v8f zero8() {
  v8f z;
#pragma unroll
  for (int i = 0; i < 8; ++i) z[i] = 0.0f;
  return z;
}
// A[M=frame][K] from LDS (row stride `st`), per-lane layout of 16x4 f32 A
__device__ __forceinline__ v2f loadA(const float* arr, int st, int k0, int lane) {
  int half = lane >> 4, fm = lane & 15;
  int k = k0 + 2 * half;
  v2f a;
  a.x = arr[fm * st + k];
  a.y = arr[fm * st + k + 1];
  return a;
}
// B[K][N] from global row-major (row stride bst), 4x16 f32 B
__device__ __forceinline__ v2f loadB(const float* mat, int bst, int k0, int nb, int lane) {
  int half = lane >> 4, col = lane & 15;
  int k = k0 + 2 * half;
  v2f b;
  b.x = mat[k * bst + nb + col];
  b.y = mat[(k + 1) * bst + nb + col];
  return b;
}
// store 16x16 D tile to LDS matrix [frame][col], cols < colmax
__device__ __forceinline__ void storeD(float* dst, int st, int nb, int colmax, v8f acc, int lane) {
  int half = lane >> 4, col = lane & 15;
  int n = nb + col;
  if (n < colmax) {
#pragma unroll
    for (int r = 0; r < 8; ++r) dst[(r + 8 * half) * st + n] = acc[r];
  }
}

// ---------------- prep kernels ----------------
__device__ static void cft_serial(float* A, int cols, int rows, float alpha) {
  float beta = 1.0f - alpha * alpha;
  for (int i = 0; i < rows * cols; ++i) A[i] = 0.0f;
  A[0] = 1.0f;
  if (rows > 1 && cols > 1) {
    float p = beta;
    for (int j = 1; j < cols; ++j) { A[cols + j] = p; p *= alpha; }
  }
  for (int i = 2; i < rows; ++i)
    for (int j = 1; j < cols; ++j)
      A[i * cols + j] = A[(i - 1) * cols + (j - 1)] +
                        alpha * (A[i * cols + (j - 1)] - A[(i - 1) * cols + j]);
}

__global__ void prep_cft_kernel(float* ws) {
  int t = threadIdx.x;
  if (t == 0) cft_serial(ws + OFF_AC, MM + 1, LL, -AL);   // AC: (512,25), alpha=-0.42
  else if (t == 1) cft_serial(ws + OFF_AP, LL, NP, AL);   // AP: (49,512)
  else if (t == 2) cft_serial(ws + OFF_AR, LL, NR, AL);   // AR: (25,512)
  if (t < LL) {
    float ang = 6.283185307179586f * (float)t / (float)LL;
    ws[OFF_TABC + t] = cosf(ang);
    ws[OFF_TABS + t] = sinf(ang);
  }
}

__global__ void prep_mat_kernel(float* ws) {
  const int E_M = MM * HPAD;       // 6528
  const int E_P = HPAD * NPAD;     // 17408
  const int E_R = HPAD * NRPAD;    // 8704
  int idx = blockIdx.x * blockDim.x + threadIdx.x;
  if (idx >= 2 * E_M + 2 * E_P + 2 * E_R) return;
  const float* tabc = ws + OFF_TABC;
  const float* tabs = ws + OFF_TABS;
  const float* AC = ws + OFF_AC;
  const float* AP = ws + OFF_AP;
  const float* AR = ws + OFF_AR;
  float val = 0.0f;
  int dst;
  if (idx < 2 * E_M) {                       // MR / MI : [24][272]
    int im = (idx >= E_M);
    int e = idx - im * E_M;
    int j = e / HPAD, k = e % HPAD;
    dst = (im ? OFF_MI : OFF_MR) + e;
    if (k < NH1) {
      float acc = 0.0f;
      const float* tab = im ? tabs : tabc;
      for (int n = 0; n < LL; ++n)
        acc += AC[n * (MM + 1) + (j + 1)] * tab[(k * n) & (LL - 1)];
      val = im ? -acc : acc;
    }
  } else if (idx < 2 * E_M + 2 * E_P) {      // PReal / PImag : [272][64]
    int e0 = idx - 2 * E_M;
    int im = (e0 >= E_P);
    int e = e0 - im * E_P;
    int k = e / NPAD, m = e % NPAD;
    dst = (im ? OFF_PI : OFF_PR) + e;
    if (k < NH1 && m < NP) {
      float w = (k == 0 || k == LL / 2) ? 1.0f : 2.0f;
      float acc = 0.0f;
      const float* tab = im ? tabs : tabc;
      for (int n = 0; n < LL; ++n)
        acc += AP[m * LL + n] * tab[(k * n) & (LL - 1)];
      val = (im ? -w : w) / (float)LL * acc;
    }
  } else {                                   // RReal / RImag : [272][32]
    int e0 = idx - 2 * E_M - 2 * E_P;
    int im = (e0 >= E_R);
    int e = e0 - im * E_R;
    int k = e / NRPAD, m = e % NRPAD;
    dst = (im ? OFF_RI : OFF_RR) + e;
    if (k < NH1 && m < NR) {
      float w = (k == 0 || k == LL / 2) ? 1.0f : 2.0f;
      float acc = 0.0f;
      const float* tab = im ? tabs : tabc;
      for (int n = 0; n < LL; ++n)
        acc += AR[m * LL + n] * tab[(k * n) & (LL - 1)];
      val = (im ? -w : w) / (float)LL * acc;
    }
  }
  ws[dst] = val;
}

// ---------------- per-frame 24x24 solve (Gaussian elim, no pivot) ----------------
__device__ static void solve24(float* sMat, float* sDel, int lane) {
  int f = lane & 15, half = lane >> 4;
  for (int k = 0; k < 23; ++k) {
    __syncthreads();
    float inv = 1.0f / sMat[(f * 24 + k) * 25 + k];
    const float* rowk = &sMat[(f * 24 + k) * 25];
    for (int i = k + 1 + half; i < 24; i += 2) {
      float* rowi = &sMat[(f * 24 + i) * 25];
      float fac = rowi[k] * inv;
      for (int j = k; j <= 24; ++j) rowi[j] -= fac * rowk[j];
    }
  }
  __syncthreads();
  if (lane < 16) {
    int ff = lane;
    for (int i = 23; i >= 0; --i) {
      float s = sMat[(ff * 24 + i) * 25 + 24];
      for (int j = i + 1; j < 24; ++j) s -= sMat[(ff * 24 + i) * 25 + j] * sDel[ff * 24 + j];
      sDel[ff * 24 + i] = s / sMat[(ff * 24 + i) * 25 + i];
    }
  }
  __syncthreads();
}

// LDS budget: sX + union(sXr|sYi , sMat) + sP/sQ/sPt/sQt + sR + sB1/sDel + sB0 + sCa/sCb
// union region: max(2*16*HPAD, 16*24*25) = 9600 floats.
// Total = 4352 + 9600 + 3328 + 448 + 768 + 16 + 832 = 19344 floats (~77 KB)
// -> 4 single-wave workgroups per 320 KB WGP (one per SIMD32).
#define UNI_FLOATS  (16*24*25)
#define SMEM_FLOATS (16*HPAD + UNI_FLOATS + 16*52*4 + 16*28 + 16*24*2 + 16 + 16*26*2)

// ---------------- main kernel: one wave32 per 16-frame tile ----------------
__global__ void __launch_bounds__(32)
mgcep_main(const float* __restrict__ x, float* __restrict__ out,
           const float* __restrict__ ws, int T) {
  extern __shared__ float sm[];
  float* sX   = sm;                    // 16*272, live whole kernel
  float* uni  = sX + 16 * HPAD;        // overlay region (9600 floats)
  float* sXr  = uni;                   // stage A/B scratch  (lifetime A->B)
  float* sYi  = uni + 16 * HPAD;
  float* sMat = uni;                   // solver scratch     (lifetime build->backsub)
  float* sP   = uni  + UNI_FLOATS;     // 16*52
  float* sQ   = sP   + 16 * 52;
  float* sPt  = sQ   + 16 * 52;
  float* sQt  = sPt  + 16 * 52;
  float* sR   = sQt  + 16 * 52;        // 16*28
  float* sB1  = sR   + 16 * 28;        // 16*24
  float* sDel = sB1  + 16 * 24;
  float* sB0  = sDel + 16 * 24;        // 16
  float* sCa  = sB0  + 16;             // 16*26
  float* sCb  = sCa  + 16 * 26;

  const float* MR = ws + OFF_MR;
  const float* MI = ws + OFF_MI;
  const float* PR = ws + OFF_PR;
  const float* PIm = ws + OFF_PI;
  const float* RRe = ws + OFF_RR;
  const float* RIm = ws + OFF_RI;

  int lane = threadIdx.x;
  int f0 = blockIdx.x * 16;

  // warm caches: one prefetch per 128B line of the shared constant matrices
  {
    const float* cbase = ws + OFF_MR;
    for (int i = lane * 32; i < CONST_FLOATS; i += 32 * 32)
      __builtin_prefetch(cbase + i, 0, 1);
  }

  // load spectrum tile (zero-padded bins)
  for (int idx = lane; idx < 16 * HPAD; idx += 32) {
    int f = idx / HPAD, k = idx % HPAD;
    int fg = f0 + f; if (fg >= T) fg = T - 1;
    sX[f * HPAD + k] = (k < NH1) ? x[fg * NH1 + k] : 0.0f;
  }
  __syncthreads();

  // ---- forward init: p0 = x @ PReal ----
  {
    v8f accP[4];
#pragma unroll
    for (int t = 0; t < 4; ++t) accP[t] = zero8();
    for (int kc = 0; kc < HPAD / 4; ++kc) {
      int k0 = kc * 4;
      v2f a = loadA(sX, HPAD, k0, lane);
#pragma unroll
      for (int t = 0; t < 4; ++t) {
        v2f b = loadB(PR, NPAD, k0, t * 16, lane);
        accP[t] = wmma4(a, b, accP[t]);
      }
    }
#pragma unroll
    for (int t = 0; t < 4; ++t) storeD(sP, 52, t * 16, 52, accP[t], lane);
  }
  __syncthreads();
  // r = p0[:25]
  for (int idx = lane; idx < 16 * NR; idx += 32) {
    int f = idx / NR, j = idx % NR;
    sR[f * 28 + j] = sP[f * 52 + j];
  }
  // p = p0 @ PT_T (closed-form bidiagonal)
  for (int idx = lane; idx < 16 * NP; idx += 32) {
    int f = idx / NP, j = idx % NP;
    float pj = sP[f * 52 + j];
    float pj1 = (j + 1 < NP) ? sP[f * 52 + j + 1] : 0.0f;
    float v;
    if (j == 0)           v = (1.0f - AL * AL) * pj + 2.0f * AL * pj1;
    else if (j == NP - 1) v = (1.0f + AL) * pj;
    else                  v = pj + AL * pj1;
    sPt[f * 52 + j] = v;
  }
  __syncthreads();
  // Toeplitz system R b1 = r[1:]
  for (int idx = lane; idx < 16 * 24 * 25; idx += 32) {
    int f = idx / 600, rem = idx % 600, i = rem / 25, j = rem % 25;
    sMat[idx] = (j < 24) ? sPt[f * 52 + ((i > j) ? (i - j) : (j - i))]
                         : sR[f * 28 + i + 1];
  }
  __syncthreads();
  solve24(sMat, sDel, lane);
  for (int idx = lane; idx < 16 * 24; idx += 32) sB1[idx] = sDel[idx];
  __syncthreads();
  if (lane < 16) {
    int f = lane;
    float acc = 0.0f;
    for (int j = 0; j < 24; ++j) acc += sR[f * 28 + 1 + j] * sB1[f * 24 + j];
    float eps = sR[f * 28] - acc;
    sB0[f] = sqrtf(fmaxf(eps, 1e-30f));
  }
  __syncthreads();

  // ---- gamma conversion chain (per-frame serial, lane f < 16) ----
  if (lane < 16) {
    int f = lane;
    // v = ignorm(b, -1)
    float K = 1.0f / sB0[f];
    sCa[f * 26 + 0] = 1.0f - K;
    for (int j = 1; j <= 24; ++j) sCa[f * 26 + j] = sB1[f * 24 + j - 1] * K;
    sCa[f * 26 + 25] = 0.0f;
    // c = v @ B2MC_T (upper-bidiagonal)
    for (int j = 0; j <= 24; ++j)
      sCb[f * 26 + j] = sCa[f * 26 + j] + AL * sCa[f * 26 + j + 1];
    // c1 = gnorm(c, -1)
    float invz = 1.0f / (1.0f - sCb[f * 26]);
    sCa[f * 26] = invz;
    for (int j = 1; j <= 24; ++j) sCa[f * 26 + j] = sCb[f * 26 + j] * invz;
    // c2 = gc2gc(c1, -1, GAM)
    sCb[f * 26 + 0] = sCa[f * 26 + 0];
    sCb[f * 26 + 1] = sCa[f * 26 + 1];
    for (int i = 2; i <= 24; ++i) {
      float ss1 = 0.0f, ss2 = 0.0f;
      for (int k = 1; k < i; ++k) {
        float cc = sCa[f * 26 + k] * sCb[f * 26 + (i - k)];
        ss2 += (float)k * cc;
        ss1 += (float)(i - k) * cc;
      }
      sCb[f * 26 + i] = sCa[f * 26 + i] + (GAM * ss2 + ss1) / (float)i;  // g1=-1
    }
    // c3 = ignorm(c2, GAM): K2 = c2[0]^(-0.5)
    float K2 = 1.0f / sqrtf(fmaxf(sCb[f * 26], 1e-30f));
    sCb[f * 26] = (K2 - 1.0f) / GAM;
    for (int j = 1; j <= 24; ++j) sCb[f * 26 + j] *= K2;
    // cb = c3 @ MC2B_T (reverse geometric recurrence, ratio -AL)
    float s = 0.0f;
    for (int j = 24; j >= 0; --j) { s = sCb[f * 26 + j] + (-AL) * s; sCb[f * 26 + j] = s; }
    // b1 = gnorm(cb, GAM)[1:]
    float invz2 = 1.0f / (1.0f + GAM * sCb[f * 26]);
    for (int j = 0; j < 24; ++j) sB1[f * 24 + j] = sCb[f * 26 + j + 1] * invz2;
  }
  __syncthreads();

  // ---- Newton iterations ----
  for (int it = 0; it < NIT; ++it) {
    // Stage A: Xr = 1 + g*(b1 @ MR), Yi = g*(b1 @ MI)   [16 frames x 272 bins]
    for (int t = 0; t < HPAD / 16; ++t) {
      v8f aX = zero8(), aY = zero8();
      for (int kc = 0; kc < 6; ++kc) {
        int k0 = kc * 4;
        v2f a = loadA(sB1, 24, k0, lane);
        v2f bR = loadB(MR, HPAD, k0, t * 16, lane);
        aX = wmma4(a, bR, aX);
        v2f bI = loadB(MI, HPAD, k0, t * 16, lane);
        aY = wmma4(a, bI, aY);
      }
      int half = lane >> 4, col = lane & 15;
      int n = t * 16 + col;
#pragma unroll
      for (int r = 0; r < 8; ++r) {
        sXr[(r + 8 * half) * HPAD + n] = 1.0f + GAM * aX[r];
        sYi[(r + 8 * half) * HPAD + n] = GAM * aY[r];
      }
    }
    __syncthreads();

    // Stage B: p/q/r matmuls with fused elementwise (ps = x*D, qs = x since -1/g==2)
    v8f accP[4], accQ[4], accR2[2];
#pragma unroll
    for (int t = 0; t < 4; ++t) { accP[t] = zero8(); accQ[t] = zero8(); }
#pragma unroll
    for (int t = 0; t < 2; ++t) accR2[t] = zero8();
    for (int kc = 0; kc < HPAD / 4; ++kc) {
      int k0 = kc * 4;
      int half = lane >> 4, fm = lane & 15;
      int kb0 = k0 + 2 * half;
      float d_p[2], d_qr[2], d_qi[2], d_rr[2], d_ri[2];
#pragma unroll
      for (int u = 0; u < 2; ++u) {
        int kb = kb0 + u;
        float xv = sX[fm * HPAD + kb];
        float Xr = sXr[fm * HPAD + kb];
        float Yi = sYi[fm * HPAD + kb];
        float XX = Xr * Xr, YY = Yi * Yi, D = XX + YY;
        float xD = xv * D;
        d_p[u]  = xD;
        d_qr[u] = xv * (XX - YY);
        d_qi[u] = xv * (2.0f * Xr * Yi);
        d_rr[u] = xD * Xr;
        d_ri[u] = xD * Yi;
      }
      v2f aP  = {d_p[0],  d_p[1]};
      v2f aQr = {d_qr[0], d_qr[1]};
      v2f aQi = {d_qi[0], d_qi[1]};
      v2f aRr = {d_rr[0], d_rr[1]};
      v2f aRi = {d_ri[0], d_ri[1]};
#pragma unroll
      for (int t = 0; t < 4; ++t) {
        v2f bPr = loadB(PR, NPAD, k0, t * 16, lane);
        accP[t] = wmma4(aP, bPr, accP[t]);
        accQ[t] = wmma4(aQr, bPr, accQ[t]);
        v2f bPi = loadB(PIm, NPAD, k0, t * 16, lane);
        accQ[t] = wmma4(aQi, bPi, accQ[t]);
      }
#pragma unroll
      for (int t = 0; t < 2; ++t) {
        v2f bRr = loadB(RRe, NRPAD, k0, t * 16, lane);
        accR2[t] = wmma4(aRr, bRr, accR2[t]);
        v2f bRi = loadB(RIm, NRPAD, k0, t * 16, lane);
        accR2[t] = wmma4(aRi, bRi, accR2[t]);
      }
    }
#pragma unroll
    for (int t = 0; t < 4; ++t) {
      storeD(sP, 52, t * 16, 52, accP[t], lane);
      storeD(sQ, 52, t * 16, 52, accQ[t], lane);
    }
#pragma unroll
    for (int t = 0; t < 2; ++t) storeD(sR, 28, t * 16, 28, accR2[t], lane);
    __syncthreads();

    // p@PT_T, q@QT_T closed forms
    for (int idx = lane; idx < 16 * NP; idx += 32) {
      int f = idx / NP, j = idx % NP;
      float pj = sP[f * 52 + j];
      float pj1 = (j + 1 < NP) ? sP[f * 52 + j + 1] : 0.0f;
      float v;
      if (j == 0)           v = (1.0f - AL * AL) * pj + 2.0f * AL * pj1;
      else if (j == NP - 1) v = (1.0f + AL) * pj;
      else                  v = pj + AL * pj1;
      sPt[f * 52 + j] = v;
      float qj = sQ[f * 52 + j];
      float qm = (j > 0) ? sQ[f * 52 + j - 1] : 0.0f;
      float w;
      if (j == 0)      w = qj;
      else if (j == 1) w = (1.0f + AL) * qj;
      else             w = qj + AL * qm;
      sQt[f * 52 + j] = w;
    }
    __syncthreads();
    // eps uses pre-update b1
    if (lane < 16) {
      int f = lane;
      float acc = 0.0f;
      for (int j = 1; j <= 24; ++j) acc += sR[f * 28 + j] * sB1[f * 24 + j - 1];
      float eps = sR[f * 28] + GAM * acc;
      sB0[f] = sqrtf(fmaxf(eps, 1e-30f));
    }
    __syncthreads();   // sXr/sYi fully consumed; safe to overlay sMat
    // (R + Q) delta = r[1:]
    for (int idx = lane; idx < 16 * 24 * 25; idx += 32) {
      int f = idx / 600, rem = idx % 600, i = rem / 25, j = rem % 25;
      sMat[idx] = (j < 24)
          ? sPt[f * 52 + ((i > j) ? (i - j) : (j - i))] +
            (1.0f + GAM) * sQt[f * 52 + i + j + 2]
          : sR[f * 28 + i + 1];
    }
    __syncthreads();
    solve24(sMat, sDel, lane);
    for (int idx = lane; idx < 16 * 24; idx += 32) sB1[idx] += sDel[idx];
    __syncthreads();
  }

  // ---- output: ignorm([b0,b1], GAM) @ B2MC_T ----
  for (int idx = lane; idx < 16 * NR; idx += 32) {
    int f = idx / NR, j = idx % NR;
    int fg = f0 + f;
    if (fg < T) {
      float K = 1.0f / sqrtf(fmaxf(sB0[f], 1e-30f));   // b0^GAM
      float vj  = (j == 0) ? (K - 1.0f) / GAM : sB1[f * 24 + j - 1] * K;
      float vj1 = (j < 24) ? sB1[f * 24 + j] * K : 0.0f;
      out[fg * NR + j] = vj + AL * vj1;
    }
  }
}

// ---------------- launch ----------------
extern "C" void kernel_launch(void* const* d_in, const int* in_sizes, int n_in,
                              void* d_out, int out_size, void* d_ws, size_t ws_size,
                              hipStream_t stream) {
  (void)n_in; (void)out_size; (void)ws_size;
  const float* x = (const float*)d_in[0];
  float* out = (float*)d_out;
  float* ws = (float*)d_ws;
  int T = in_sizes[0] / NH1;

  prep_cft_kernel<<<1, 512, 0, stream>>>(ws);
  int E = 2 * MM * HPAD + 2 * HPAD * NPAD + 2 * HPAD * NRPAD;
  prep_mat_kernel<<<(E + 255) / 256, 256, 0, stream>>>(ws);

  int tiles = (T + 15) / 16;
  size_t lds = (size_t)SMEM_FLOATS * sizeof(float);
  mgcep_main<<<tiles, 32, lds, stream>>>(x, out, ws, T);
}